// DiceLoss_49778670961267
// MI455X (gfx1250) — compile-verified
//
#include <hip/hip_runtime.h>

// DiceLoss partial-sums on gfx1250 (MI455X): bandwidth-bound dual reduction.
// 268 MB streamed once -> ~11.5 us floor at 23.3 TB/s.
// Cross-lane reductions use V_WMMA_F32_16X16X4_F32 (exact f32, RNE).

typedef __attribute__((ext_vector_type(2))) float v2f;
typedef __attribute__((ext_vector_type(4))) float f32x4;
typedef __attribute__((ext_vector_type(8))) float v8f;

#define NTHR 256
#define MAX_BLOCKS 2048
#define NWAVES (NTHR / 32)

// Exact 32-lane f32 sum via one WMMA + one xor-16 shuffle.
// A = {v, 0} per lane -> A[m][0]=v_m, A[m][2]=v_{m+16}, A[m][1]=A[m][3]=0.
// B = all ones       -> D[m][n] = v_m + v_{m+16}  (every column identical).
// Lane (l<16) sums its 8 D VGPRs = sum_{m=0..7}(v_m+v_{m+16});
// lane (l>=16) = sum_{m=8..15}(...); xor-16 fold completes the wave sum.
// EXEC must be all 1s at the call site (callers guarantee reconvergence).
__device__ __forceinline__ float wave_sum32(float v) {
    v2f a; a.x = v;    a.y = 0.0f;
    v2f b; b.x = 1.0f; b.y = 1.0f;
    v8f c = {};
    v8f d = __builtin_amdgcn_wmma_f32_16x16x4_f32(
        /*neg_a=*/false, a, /*neg_b=*/false, b,
        /*c_mod=*/(short)0, c, /*reuse_a=*/false, /*reuse_b=*/false);
    float t = ((d[0] + d[1]) + (d[2] + d[3])) + ((d[4] + d[5]) + (d[6] + d[7]));
    t += __shfl_xor(t, 16, 32);
    return t;
}

__global__ __launch_bounds__(NTHR)
void dice_partial_kernel(const float* __restrict__ in,
                         const float* __restrict__ tgt,
                         float* __restrict__ ws,
                         long long n, int nblk) {
    const long long tid     = (long long)blockIdx.x * NTHR + threadIdx.x;
    const long long nthread = (long long)gridDim.x * NTHR;
    const long long n4      = n >> 2;

    const f32x4* __restrict__ in4 = (const f32x4*)in;
    const f32x4* __restrict__ tg4 = (const f32x4*)tgt;

    float s_int = 0.0f, s_bin = 0.0f, s_tgt = 0.0f;

    // Streaming loop: b128 non-temporal loads, branch-free binarize.
    #pragma unroll 4
    for (long long i = tid; i < n4; i += nthread) {
        f32x4 x = __builtin_nontemporal_load(in4 + i);
        f32x4 t = __builtin_nontemporal_load(tg4 + i);
        #pragma unroll
        for (int c = 0; c < 4; ++c) {
            bool gt = x[c] > 0.5f;            // strictly greater (Binarizer)
            s_bin += gt ? 1.0f : 0.0f;
            s_int += gt ? t[c] : 0.0f;
            s_tgt += t[c];
        }
    }
    // Scalar tail (empty for N = 32*1024*1024; reconverges before WMMA).
    if (tid == 0) {
        for (long long i = n4 << 2; i < n; ++i) {
            bool gt = in[i] > 0.5f;
            s_bin += gt ? 1.0f : 0.0f;
            s_int += gt ? tgt[i] : 0.0f;
            s_tgt += tgt[i];
        }
    }

    // Wave-level exact reduction via WMMA (EXEC all-ones here).
    s_int = wave_sum32(s_int);
    s_bin = wave_sum32(s_bin);
    s_tgt = wave_sum32(s_tgt);

    __shared__ float l_int[NWAVES], l_bin[NWAVES], l_tgt[NWAVES];
    const int wave = threadIdx.x >> 5;
    const int lane = threadIdx.x & 31;
    if (lane == 0) { l_int[wave] = s_int; l_bin[wave] = s_bin; l_tgt[wave] = s_tgt; }
    __syncthreads();

    if (threadIdx.x == 0) {
        float a = 0.0f, b = 0.0f, c = 0.0f;
        #pragma unroll
        for (int w = 0; w < NWAVES; ++w) { a += l_int[w]; b += l_bin[w]; c += l_tgt[w]; }
        ws[blockIdx.x]            = a;   // intersection partial
        ws[nblk + blockIdx.x]     = b;   // bin-count partial
        ws[2 * nblk + blockIdx.x] = c;   // target-sum partial
    }
}

__global__ __launch_bounds__(NTHR)
void dice_final_kernel(const float* __restrict__ ws,
                       float* __restrict__ out, int nblk) {
    float s_int = 0.0f, s_bin = 0.0f, s_tgt = 0.0f;
    for (int i = threadIdx.x; i < nblk; i += NTHR) {
        s_int += ws[i];
        s_bin += ws[nblk + i];
        s_tgt += ws[2 * nblk + i];
    }

    s_int = wave_sum32(s_int);
    s_bin = wave_sum32(s_bin);
    s_tgt = wave_sum32(s_tgt);

    __shared__ float l_int[NWAVES], l_bin[NWAVES], l_tgt[NWAVES];
    const int wave = threadIdx.x >> 5;
    const int lane = threadIdx.x & 31;
    if (lane == 0) { l_int[wave] = s_int; l_bin[wave] = s_bin; l_tgt[wave] = s_tgt; }
    __syncthreads();

    if (threadIdx.x == 0) {
        float a = 0.0f, b = 0.0f, c = 0.0f;
        #pragma unroll
        for (int w = 0; w < NWAVES; ++w) { a += l_int[w]; b += l_bin[w]; c += l_tgt[w]; }
        out[0] = 2.0f * a;   // loss1 = 2 * intersection
        out[1] = b + c;      // loss2 = bin.sum() + target.sum()
    }
}

extern "C" void kernel_launch(void* const* d_in, const int* in_sizes, int n_in,
                              void* d_out, int out_size, void* d_ws, size_t ws_size,
                              hipStream_t stream) {
    const float* in  = (const float*)d_in[0];   // "input"
    const float* tgt = (const float*)d_in[1];   // "target"
    float* out = (float*)d_out;                 // 2 f32 scalars
    float* ws  = (float*)d_ws;

    const long long n = (long long)in_sizes[0];

    int nblk = MAX_BLOCKS;                      // needs 3*nblk*4 B of scratch (24 KB)
    const size_t need_per_blk = 3 * sizeof(float);
    if (ws_size < (size_t)nblk * need_per_blk) {
        nblk = (int)(ws_size / need_per_blk);
        if (nblk < 1) nblk = 1;
    }

    dice_partial_kernel<<<nblk, NTHR, 0, stream>>>(in, tgt, ws, n, nblk);
    dice_final_kernel<<<1, NTHR, 0, stream>>>(ws, out, nblk);
}